// NASA_loss_9929964389095
// MI455X (gfx1250) — compile-verified
//
#include <hip/hip_runtime.h>
#include <math.h>

// MI455X / gfx1250: wave32, V_WMMA_F32_16X16X4_F32 for fp32-exact GEMM tiles.

typedef __attribute__((ext_vector_type(2))) float v2f;
typedef __attribute__((ext_vector_type(8))) float v8f;

#define M_DIM 1024
#define D_DIM 128
#define NT    (M_DIM / 16)   // 64 tiles per side
#define NTILES (NT * NT)     // 4096
#define MARGIN 3.0f
#define SCALE_A 10.0f
#define EPS_F 1e-12f

// ---------------------------------------------------------------------------
// Kernel 0: per-row sums  sq[i] = sum_d e^2,  asum[i] = sum_d strc*e^2
// ---------------------------------------------------------------------------
__global__ void row_stats_kernel(const float* __restrict__ E,
                                 const float* __restrict__ S,
                                 float* __restrict__ sq,
                                 float* __restrict__ asum) {
    int i = blockIdx.x * blockDim.x + threadIdx.x;
    if (i >= M_DIM) return;
    const float* e = E + i * D_DIM;
    const float* s = S + i * D_DIM;
    float a = 0.f, b = 0.f;
#pragma unroll 8
    for (int d = 0; d < D_DIM; ++d) {
        float x = e[d];
        a += x * x;
        b += s[d] * x * x;
    }
    sq[i] = a;
    asum[i] = b;
}

// ---------------------------------------------------------------------------
// Kernel 1: one wave32 per 16x16 tile of prod = E*E^T via V_WMMA_F32_16X16X4_F32.
// Epilogue forms dist entries, zeroes the diagonal, wave-reduces (sum, sum^2).
// ---------------------------------------------------------------------------
__global__ __launch_bounds__(32)
void dist_tile_kernel(const float* __restrict__ E,
                      const float* __restrict__ sq,
                      float* __restrict__ part1,
                      float* __restrict__ part2) {
    const int ti = blockIdx.y, tj = blockIdx.x;
    const int i0 = ti * 16, j0 = tj * 16;
    const int lane = threadIdx.x;
    const int r = lane & 15;       // M (A) / N (B,C,D) index
    const int h = lane >> 4;       // K-pair selector (A/B), +8 row selector (C/D)

    // A lane holds E[i0+r][k + 2h .. k+2h+1]; B lane holds E[j0+r][k + 2h ..]
    const float* pa = E + (i0 + r) * D_DIM + 2 * h;
    const float* pb = E + (j0 + r) * D_DIM + 2 * h;

    v8f acc = {};
#pragma unroll 4
    for (int k = 0; k < D_DIM; k += 4) {
        v2f a = *(const v2f*)(pa + k);
        v2f b = *(const v2f*)(pb + k);
        acc = __builtin_amdgcn_wmma_f32_16x16x4_f32(
            /*neg_a=*/false, a, /*neg_b=*/false, b,
            /*c_mod=*/(short)0, acc, /*reuse_a=*/false, /*reuse_b=*/false);
    }

    // D layout: lane (r, h) holds rows m = c + 8h (c = 0..7), column n = r.
    const int j = j0 + r;
    const float sqj = sq[j];
    float s1 = 0.f, s2 = 0.f;
#pragma unroll
    for (int c = 0; c < 8; ++c) {
        int i = i0 + c + 8 * h;
        float d2 = sq[i] + sqj - 2.0f * acc[c];
        d2 = fmaxf(d2, EPS_F);
        float dd = (i == j) ? 0.0f : sqrtf(d2);
        s1 += dd;
        s2 += dd * dd;
    }
    // deterministic wave32 butterfly reduction
    for (int off = 16; off >= 1; off >>= 1) {
        s1 += __shfl_xor(s1, off, 32);
        s2 += __shfl_xor(s2, off, 32);
    }
    if (lane == 0) {
        int tile = ti * NT + tj;
        part1[tile] = s1;
        part2[tile] = s2;
    }
}

// ---------------------------------------------------------------------------
// Kernel 2: reduce 4096 tile partials -> mean/std (ddof=1) -> eta0, eta1
// ---------------------------------------------------------------------------
__global__ void stats_kernel(const float* __restrict__ part1,
                             const float* __restrict__ part2,
                             float* __restrict__ eta) {
    __shared__ float sm1[256];
    __shared__ float sm2[256];
    float a = 0.f, b = 0.f;
    for (int t = threadIdx.x; t < NTILES; t += 256) {
        a += part1[t];
        b += part2[t];
    }
    sm1[threadIdx.x] = a;
    sm2[threadIdx.x] = b;
    __syncthreads();
    for (int off = 128; off > 0; off >>= 1) {
        if (threadIdx.x < off) {
            sm1[threadIdx.x] += sm1[threadIdx.x + off];
            sm2[threadIdx.x] += sm2[threadIdx.x + off];
        }
        __syncthreads();
    }
    if (threadIdx.x == 0) {
        const float N = (float)M_DIM * (float)M_DIM;   // flat includes zero diagonal
        float S1 = sm1[0], S2 = sm2[0];
        float mean = S1 / N;
        float var = (S2 - S1 * S1 / N) / (N - 1.0f);   // unbiased (torch std ddof=1)
        float sd = sqrtf(fmaxf(var, 0.0f));
        eta[0] = mean;                                  // eta0
        eta[1] = fmaxf(0.2f * mean, mean - MARGIN * sd); // eta1
    }
}

// ---------------------------------------------------------------------------
// Kernel 3: per tile, two WMMA accumulators:
//   P = (strc .* E) * E^T,   Q = strc * (E .* E)^T
// adapt^2 = asum[i] - 2P + Q ; off-diagonal soft-margin loss; wave-reduce.
// ---------------------------------------------------------------------------
__global__ __launch_bounds__(32)
void loss_tile_kernel(const float* __restrict__ E,
                      const float* __restrict__ S,
                      const float* __restrict__ asum,
                      const float* __restrict__ eta,
                      float* __restrict__ partL) {
    const int ti = blockIdx.y, tj = blockIdx.x;
    const int i0 = ti * 16, j0 = tj * 16;
    const int lane = threadIdx.x;
    const int r = lane & 15;
    const int h = lane >> 4;

    const float* pe_i = E + (i0 + r) * D_DIM + 2 * h;
    const float* ps_i = S + (i0 + r) * D_DIM + 2 * h;
    const float* pe_j = E + (j0 + r) * D_DIM + 2 * h;

    v8f accP = {};
    v8f accQ = {};
#pragma unroll 2
    for (int k = 0; k < D_DIM; k += 4) {
        v2f ei = *(const v2f*)(pe_i + k);
        v2f si = *(const v2f*)(ps_i + k);
        v2f ej = *(const v2f*)(pe_j + k);
        v2f se = si * ei;   // (strc .* E) row i, in registers
        v2f ee = ej * ej;   // (E .* E) row j, in registers
        accP = __builtin_amdgcn_wmma_f32_16x16x4_f32(
            false, se, false, ej, (short)0, accP, false, false);
        accQ = __builtin_amdgcn_wmma_f32_16x16x4_f32(
            false, si, false, ee, (short)0, accQ, false, false);
    }

    const float eta0 = eta[0];
    const float eta1 = eta[1];
    const int j = j0 + r;
    float sL = 0.f;
#pragma unroll
    for (int c = 0; c < 8; ++c) {
        int i = i0 + c + 8 * h;
        if (i == j) continue;   // _remove_diag
        float a2 = asum[i] - 2.0f * accP[c] + accQ[c];
        a2 = fmaxf(a2, 0.0f);
        float ad = sqrtf(a2);
        float pos = fabsf(ad - eta0);
        float neg = fabsf(ad - eta1);
        float x = SCALE_A * (neg - pos);
        float cc = 1.0f / (1.0f + __expf(-x));  // sigmoid
        sL += cc * pos + (1.0f - cc) * neg;
    }
    for (int off = 16; off >= 1; off >>= 1)
        sL += __shfl_xor(sL, off, 32);
    if (lane == 0)
        partL[ti * NT + tj] = sL;
}

// ---------------------------------------------------------------------------
// Kernel 4: reduce per-tile loss partials -> scalar mean over M*(M-1)
// ---------------------------------------------------------------------------
__global__ void finalize_kernel(const float* __restrict__ partL,
                                float* __restrict__ out) {
    __shared__ float sm[256];
    float a = 0.f;
    for (int t = threadIdx.x; t < NTILES; t += 256) a += partL[t];
    sm[threadIdx.x] = a;
    __syncthreads();
    for (int off = 128; off > 0; off >>= 1) {
        if (threadIdx.x < off) sm[threadIdx.x] += sm[threadIdx.x + off];
        __syncthreads();
    }
    if (threadIdx.x == 0)
        out[0] = sm[0] / ((float)M_DIM * (float)(M_DIM - 1));
}

// ---------------------------------------------------------------------------
extern "C" void kernel_launch(void* const* d_in, const int* in_sizes, int n_in,
                              void* d_out, int out_size, void* d_ws, size_t ws_size,
                              hipStream_t stream) {
    const float* E = (const float*)d_in[0];   // embeddings [1024,128] fp32
    const float* S = (const float*)d_in[1];   // embeddings_strc [1024,128] fp32
    float* out = (float*)d_out;               // scalar loss
    float* ws = (float*)d_ws;

    float* sq    = ws;                        // [M]
    float* asum  = sq + M_DIM;                // [M]
    float* part1 = asum + M_DIM;              // [NTILES] sum(dist)
    float* part2 = part1 + NTILES;            // [NTILES] sum(dist^2)
    float* eta   = part2 + NTILES;            // [2]
    float* partL = eta + 2;                   // [NTILES]
    // total: (2*1024 + 3*4096 + 2) * 4B ~= 57 KB of workspace

    row_stats_kernel<<<dim3((M_DIM + 255) / 256), dim3(256), 0, stream>>>(E, S, sq, asum);
    dist_tile_kernel<<<dim3(NT, NT), dim3(32), 0, stream>>>(E, sq, part1, part2);
    stats_kernel<<<dim3(1), dim3(256), 0, stream>>>(part1, part2, eta);
    loss_tile_kernel<<<dim3(NT, NT), dim3(32), 0, stream>>>(E, S, asum, eta, partL);
    finalize_kernel<<<dim3(1), dim3(256), 0, stream>>>(partL, out);
}